// GATv2Convolution_22935125361137
// MI455X (gfx1250) — compile-verified
//
#include <hip/hip_runtime.h>
#include <math.h>

#define LRELU_ALPHA 0.2f

typedef __attribute__((ext_vector_type(2))) float v2f;
typedef __attribute__((ext_vector_type(8))) float v8f;

__device__ __forceinline__ float lrelu(float x) { return x > 0.f ? x : LRELU_ALPHA * x; }

// Order-preserving float <-> uint mapping so float-max == uint-max.
__device__ __forceinline__ unsigned f2ord(float f) {
    unsigned u = __float_as_uint(f);
    return (u & 0x80000000u) ? ~u : (u | 0x80000000u);
}
__device__ __forceinline__ float ord2f(unsigned o) {
    unsigned u = (o & 0x80000000u) ? (o & 0x7fffffffu) : ~o;
    return __uint_as_float(u);
}
#define ORD_NEG_INF 0x007FFFFFu   // f2ord(-inf)

// ---------------------------------------------------------------------------
// Init: zero d_out (pooled accumulator), segmax = -inf ordinal, denom = 0.
// ---------------------------------------------------------------------------
__global__ __launch_bounds__(256) void init_buffers(float* __restrict__ out,
                                                    unsigned* __restrict__ segmax,
                                                    float* __restrict__ denom,
                                                    int n_nodes) {
    int i = blockIdx.x * blockDim.x + threadIdx.x;
    if (i < n_nodes * 64) out[i] = 0.f;
    if (i < n_nodes * 4) { segmax[i] = ORD_NEG_INF; denom[i] = 0.f; }
}

// ---------------------------------------------------------------------------
// Node GEMM on the f32 WMMA pipe: Q = X*Wq + bq, V = X*Wv + bv.
// One wave computes a 16x64 output tile via 16 k-steps x 4 col-tiles of
// V_WMMA_F32_16X16X4_F32. blockIdx.y selects Q vs V weights.
//
// A (16x4 f32) per-lane layout: lane L holds row M=L&15; VGPR0 = K=kb,
// VGPR1 = K=kb+1 with kb = 2*(L>>4). B (4x16) mirrored: VGPR0 = row kb,
// VGPR1 = row kb+1, N = L&15. D: VGPR r holds (M = r + 8*(L>>4), N = L&15).
// ---------------------------------------------------------------------------
__global__ __launch_bounds__(128) void node_gemm_wmma(
    const float* __restrict__ X,
    const float* __restrict__ Wq, const float* __restrict__ bq,
    const float* __restrict__ Wv, const float* __restrict__ bv,
    float* __restrict__ Q, float* __restrict__ V, int n_nodes) {
    const int wave = threadIdx.x >> 5;            // 0..3 (4 waves/block)
    const int lane = threadIdx.x & 31;
    const int tile = blockIdx.x * 4 + wave;       // 16-row tile
    const int row0 = tile * 16;
    if (row0 >= n_nodes) return;

    const bool  isV  = (blockIdx.y != 0);
    const float* W    = isV ? Wv : Wq;
    const float* bias = isV ? bv : bq;
    float*       Out  = isV ? V  : Q;

    const int m  = lane & 15;
    const int kb = (lane >> 4) << 1;              // 0 or 2
    const int n  = lane & 15;

    // Load the A tile (row m, all 64 K, striped per WMMA layout).
    int arow = row0 + m; if (arow >= n_nodes) arow = n_nodes - 1;   // tail clamp
    const float* xrow = X + (size_t)arow * 64 + kb;
    v2f a[16];
#pragma unroll
    for (int s = 0; s < 16; ++s) a[s] = *(const v2f*)(xrow + 4 * s);

    v8f acc[4];
    v8f zero = {};
#pragma unroll
    for (int j = 0; j < 4; ++j) acc[j] = zero;

#pragma unroll
    for (int s = 0; s < 16; ++s) {
        const int k0 = 4 * s + kb;
#pragma unroll
        for (int j = 0; j < 4; ++j) {
            v2f b;
            b.x = W[(size_t)(k0    ) * 64 + j * 16 + n];
            b.y = W[(size_t)(k0 + 1) * 64 + j * 16 + n];
            acc[j] = __builtin_amdgcn_wmma_f32_16x16x4_f32(
                false, a[s], false, b, (short)0, acc[j], false, false);
        }
    }

    const int mofs = (lane >> 4) * 8;
#pragma unroll
    for (int j = 0; j < 4; ++j) {
        const float bval = bias[j * 16 + n];
#pragma unroll
        for (int r = 0; r < 8; ++r) {
            const int row = row0 + r + mofs;
            if (row < n_nodes)
                Out[(size_t)row * 64 + j * 16 + n] = acc[j][r] + bval;
        }
    }
}

// ---------------------------------------------------------------------------
// Pass 1: one thread per (edge, head). logit = sum_c lrelu(q_t + v_s)*ak[c,h];
// store logit, atomic-max into segmax via ordinal uint atomics.
// ---------------------------------------------------------------------------
__global__ __launch_bounds__(256) void edge_logits_kernel(
    const float* __restrict__ Q, const float* __restrict__ V,
    const float* __restrict__ AK,                       // [16,4] row-major
    const int* __restrict__ esrc, const int* __restrict__ etgt,
    float* __restrict__ logits, unsigned* __restrict__ segmax, int n_edges) {
    int idx = blockIdx.x * blockDim.x + threadIdx.x;    // edge*4 + head
    if (idx >= n_edges * 4) return;
    const int e = idx >> 2, h = idx & 3;
    const int t = etgt[e], s = esrc[e];

    float ak[16];
#pragma unroll
    for (int c = 0; c < 16; ++c) ak[c] = AK[c * 4 + h]; // L1/L2-resident

    const float4* qp = (const float4*)(Q + (size_t)t * 64 + h * 16);
    const float4* vp = (const float4*)(V + (size_t)s * 64 + h * 16);
    float acc = 0.f;
#pragma unroll
    for (int g = 0; g < 4; ++g) {
        float4 q = qp[g], v = vp[g];
        acc += lrelu(q.x + v.x) * ak[4 * g + 0];
        acc += lrelu(q.y + v.y) * ak[4 * g + 1];
        acc += lrelu(q.z + v.z) * ak[4 * g + 2];
        acc += lrelu(q.w + v.w) * ak[4 * g + 3];
    }
    logits[idx] = acc;
    atomicMax(&segmax[(size_t)t * 4 + h], f2ord(acc));
}

// ---------------------------------------------------------------------------
// Pass 2: w = exp(logit - segmax[t,h]); denom[t,h] += w;
// out[t, h*16+c] += w * v[s, h*16+c]   (unnormalized — denom divides later).
// ---------------------------------------------------------------------------
__global__ __launch_bounds__(256) void edge_scatter_kernel(
    const float* __restrict__ V,
    const int* __restrict__ esrc, const int* __restrict__ etgt,
    const float* __restrict__ logits, const unsigned* __restrict__ segmax,
    float* __restrict__ out, float* __restrict__ denom, int n_edges) {
    int idx = blockIdx.x * blockDim.x + threadIdx.x;    // edge*4 + head
    if (idx >= n_edges * 4) return;
    const int e = idx >> 2, h = idx & 3;
    const int t = etgt[e], s = esrc[e];

    float m = ord2f(segmax[(size_t)t * 4 + h]);
    if (!(m > -1e37f && m < 1e37f)) m = 0.f;            // mirror where(isfinite)
    const float w = __expf(logits[idx] - m);

    atomicAdd(&denom[(size_t)t * 4 + h], w);

    const float4* vp = (const float4*)(V + (size_t)s * 64 + h * 16);
    float* pp = out + (size_t)t * 64 + h * 16;
#pragma unroll
    for (int g = 0; g < 4; ++g) {
        float4 v = vp[g];
        atomicAdd(pp + 4 * g + 0, w * v.x);
        atomicAdd(pp + 4 * g + 1, w * v.y);
        atomicAdd(pp + 4 * g + 2, w * v.z);
        atomicAdd(pp + 4 * g + 3, w * v.w);
    }
}

// ---------------------------------------------------------------------------
// Finalize: out = leaky_relu(out / max(denom, >0 ? denom : 1)).
// ---------------------------------------------------------------------------
__global__ __launch_bounds__(256) void finalize_kernel(float* __restrict__ out,
                                                       const float* __restrict__ denom,
                                                       int n_nodes) {
    int i = blockIdx.x * blockDim.x + threadIdx.x;
    if (i >= n_nodes * 64) return;
    const int nd = i >> 6;
    const int h = (i >> 4) & 3;
    float d = denom[(size_t)nd * 4 + h];
    d = d > 0.f ? d : 1.f;
    out[i] = lrelu(out[i] / d);
}

// ---------------------------------------------------------------------------
extern "C" void kernel_launch(void* const* d_in, const int* in_sizes, int n_in,
                              void* d_out, int out_size, void* d_ws, size_t ws_size,
                              hipStream_t stream) {
    const float* X   = (const float*)d_in[0];
    const float* Wq  = (const float*)d_in[1];
    const float* bq  = (const float*)d_in[2];
    const float* Wv  = (const float*)d_in[3];
    const float* bv  = (const float*)d_in[4];
    const float* AK  = (const float*)d_in[5];
    const int*  esrc = (const int*)d_in[6];
    const int*  etgt = (const int*)d_in[7];
    float* out = (float*)d_out;

    const int n_nodes = in_sizes[0] / 64;
    const int n_edges = in_sizes[6];

    // Workspace layout (all f32/u32, generously aligned):
    float*    Q      = (float*)d_ws;                      // n_nodes*64
    float*    V      = Q + (size_t)n_nodes * 64;          // n_nodes*64
    float*    logits = V + (size_t)n_nodes * 64;          // n_edges*4
    unsigned* segmax = (unsigned*)(logits + (size_t)n_edges * 4);  // n_nodes*4
    float*    denom  = (float*)(segmax + (size_t)n_nodes * 4);     // n_nodes*4

    {   // init accumulators
        int total = n_nodes * 64;
        init_buffers<<<(total + 255) / 256, 256, 0, stream>>>(out, segmax, denom, n_nodes);
    }
    {   // WMMA node GEMMs (Q and V via blockIdx.y)
        int tiles = (n_nodes + 15) / 16;
        dim3 grid((tiles + 3) / 4, 2);
        node_gemm_wmma<<<grid, 128, 0, stream>>>(X, Wq, bq, Wv, bv, Q, V, n_nodes);
    }
    {   // edge pass 1: logits + segment max
        int total = n_edges * 4;
        edge_logits_kernel<<<(total + 255) / 256, 256, 0, stream>>>(
            Q, V, AK, esrc, etgt, logits, segmax, n_edges);
    }
    {   // edge pass 2: exp + denom + unnormalized scatter
        int total = n_edges * 4;
        edge_scatter_kernel<<<(total + 255) / 256, 256, 0, stream>>>(
            V, esrc, etgt, logits, segmax, out, denom, n_edges);
    }
    {   // normalize + final activation
        int total = n_nodes * 64;
        finalize_kernel<<<(total + 255) / 256, 256, 0, stream>>>(out, denom, n_nodes);
    }
}